// GraphFSA_22179211117296
// MI455X (gfx1250) — compile-verified
//
#include <hip/hip_runtime.h>

// GraphFSA on MI455X (gfx1250).
// Bottleneck: 6.4M-edge scatter-add (L2-resident f32 atomics). Dense-matrix
// part (per-char (n_c x 5) x (5 x 5) GEMMs) runs on V_WMMA_F32_16X16X4_F32 in
// full f32 precision (threshold dynamics are chaotic -> f16 state would
// diverge from the reference and flip characters).

#define NN      200000      // N_NODES
#define NE      6400000     // N_EDGES
#define SN      5           // STATE_N
#define CHN     1024        // CHAR_N = 4^5
#define ITERS   5

typedef __attribute__((ext_vector_type(2))) float v2f;
typedef __attribute__((ext_vector_type(8))) float v8f;

// ---------------- softmax over last axis of T[1024][5][5] ------------------
__global__ void softmax_k(const float* __restrict__ T, float* __restrict__ Ts) {
    int r = blockIdx.x * blockDim.x + threadIdx.x;     // row = (c, s)
    if (r >= CHN * SN) return;
    const float* in = T + (long)r * SN;
    float m = in[0];
#pragma unroll
    for (int i = 1; i < SN; ++i) m = fmaxf(m, in[i]);
    float e[SN], s = 0.f;
#pragma unroll
    for (int i = 0; i < SN; ++i) { e[i] = expf(in[i] - m); s += e[i]; }
    float inv = 1.f / s;
    float* outp = Ts + (long)r * SN;
#pragma unroll
    for (int i = 0; i < SN; ++i) outp[i] = e[i] * inv;
}

// ---------------- scatter-add: msum[dst] += s[src] -------------------------
__global__ void scatter_k(const int* __restrict__ src, const int* __restrict__ dst,
                          const float* __restrict__ s, float* __restrict__ msum) {
    int e = blockIdx.x * blockDim.x + threadIdx.x;
    if (e >= NE) return;
    int se = src[e], de = dst[e];
    const float* sp = s + (long)se * SN;
    float* mp = msum + (long)de * SN;
#pragma unroll
    for (int j = 0; j < SN; ++j) {
        // relaxed agent-scope fp atomic -> native global_atomic_add_f32
        __hip_atomic_fetch_add(&mp[j], sp[j], __ATOMIC_RELAXED, __HIP_MEMORY_SCOPE_AGENT);
    }
}

// ---------------- per-node character + histogram ---------------------------
__global__ void char_k(const float* __restrict__ msum, int* __restrict__ char_idx,
                       int* __restrict__ bin_cnt) {
    int n = blockIdx.x * blockDim.x + threadIdx.x;
    if (n >= NN) return;
    const float* mp = msum + (long)n * SN;
    int ch = 0, w = 1;
#pragma unroll
    for (int j = 0; j < SN; ++j) {
        float v = mp[j];
        int c = (v > 0.5f) + (v > 1.5f) + (v > 2.5f);   // already in [0,3]
        ch += c * w;
        w *= 4;
    }
    char_idx[n] = ch;
    atomicAdd(&bin_cnt[ch], 1);
}

// ---------------- exclusive scan of the 1024 bins (one block) --------------
__global__ void scan_k(const int* __restrict__ bin_cnt, int* __restrict__ bin_off,
                       int* __restrict__ bin_cur) {
    __shared__ int tmp[CHN];
    int t = threadIdx.x;
    int v0 = bin_cnt[t];
    tmp[t] = v0;
    __syncthreads();
    for (int off = 1; off < CHN; off <<= 1) {
        int v = (t >= off) ? tmp[t - off] : 0;
        __syncthreads();
        tmp[t] += v;
        __syncthreads();
    }
    bin_off[t] = tmp[t] - v0;   // exclusive
    bin_cur[t] = 0;
}

// ---------------- compact node ids into char-sorted order ------------------
__global__ void compact_k(const int* __restrict__ char_idx, const int* __restrict__ bin_off,
                          int* __restrict__ bin_cur, int* __restrict__ sorted) {
    int n = blockIdx.x * blockDim.x + threadIdx.x;
    if (n >= NN) return;
    int c = char_idx[n];
    int p = atomicAdd(&bin_cur[c], 1);
    sorted[bin_off[c] + p] = n;
}

// ---------------- grouped GEMM update: out = s . Ts[c], via WMMA -----------
// One block per character c; each wave processes 16-node tiles.
// D(16x16) = A(16x4) * B(4x16) + C ; chained twice to cover K = 5.
// Layouts (ISA 7.12.2, f32):
//   A: lanes 0-15 hold (M=lane, K=0..1), lanes 16-31 hold (M=lane-16, K=2..3)
//   B: lanes 0-15 hold (K=0..1, N=lane), lanes 16-31 hold (K=2..3, N=lane-16)
//   D: vgpr i: lanes 0-15 = (M=i, N=lane), lanes 16-31 = (M=i+8, N=lane-16)
__global__ void update_wmma_k(const float* __restrict__ s, const float* __restrict__ Ts,
                              const int* __restrict__ sorted, const int* __restrict__ bin_off,
                              const int* __restrict__ bin_cnt, float* __restrict__ out) {
    int c   = blockIdx.x;
    int cnt = bin_cnt[c];
    if (cnt == 0) return;                       // block-uniform: whole waves exit
    int base = bin_off[c];
    int wave = threadIdx.x >> 5;
    int lane = threadIdx.x & 31;
    int half = lane >> 4;       // 0: K rows {0,1}/{row4};  1: K rows {2,3}
    int nn   = lane & 15;       // M for A-load / N for B & D

    const float* Tc = Ts + (long)c * 25;        // Ts[c][k][t], row-major 5x5
    // B operands are fixed for the whole character (branchless build).
    bool n5 = nn < 5;
    float t0 = Tc[0 * 5 + (n5 ? nn : 0)];
    float t1 = Tc[1 * 5 + (n5 ? nn : 0)];
    float t2 = Tc[2 * 5 + (n5 ? nn : 0)];
    float t3 = Tc[3 * 5 + (n5 ? nn : 0)];
    float t4 = Tc[4 * 5 + (n5 ? nn : 0)];
    v2f b1, b2;
    b1.x = (n5 ? (half == 0 ? t0 : t2) : 0.f);
    b1.y = (n5 ? (half == 0 ? t1 : t3) : 0.f);
    b2.x = (n5 && half == 0) ? t4 : 0.f;
    b2.y = 0.f;

    int ntiles = (cnt + 15) >> 4;
    int nwaves = blockDim.x >> 5;
    int k0 = half * 2;
    for (int t = wave; t < ntiles; t += nwaves) {
        int tbase = base + t * 16;
        int valid = min(16, cnt - t * 16);
        bool act  = nn < valid;
        // Clamped index -> loads are always in-range; mask with cndmask after.
        int  node = sorted[tbase + (act ? nn : 0)];
        const float* sp = s + (long)node * SN;
        float v0 = sp[k0];
        float v1 = sp[k0 + 1];
        float v4 = sp[4];
        v2f a1, a2;
        a1.x = act ? v0 : 0.f;
        a1.y = act ? v1 : 0.f;
        a2.x = (act && half == 0) ? v4 : 0.f;
        a2.y = 0.f;

        v8f acc = {0.f, 0.f, 0.f, 0.f, 0.f, 0.f, 0.f, 0.f};
        acc = __builtin_amdgcn_wmma_f32_16x16x4_f32(false, a2, false, b2,
                                                    (short)0, acc, false, false);
        acc = __builtin_amdgcn_wmma_f32_16x16x4_f32(false, a1, false, b1,
                                                    (short)0, acc, false, false);
#pragma unroll
        for (int i = 0; i < 8; ++i) {
            int mo = i + half * 8;               // output row M this lane holds
            // Row-mo node id lives in lane mo (its A-row owner): wave shuffle
            // instead of 8 serialized global loads of sorted[].
            int onode = __shfl(node, mo, 32);
            if (nn < 5 && mo < valid) {
                out[(long)onode * SN + nn] = acc[i];
            }
        }
    }
}

// ---------------------------------------------------------------------------
extern "C" void kernel_launch(void* const* d_in, const int* in_sizes, int n_in,
                              void* d_out, int out_size, void* d_ws, size_t ws_size,
                              hipStream_t stream) {
    const float* s0  = (const float*)d_in[0];
    const int*   ei  = (const int*)d_in[1];      // [2, NE] int32 per harness rules
    const float* T   = (const float*)d_in[2];
    const int*   src = ei;
    const int*   dst = ei + NE;
    float* out = (float*)d_out;

    char* p = (char*)d_ws;
    float* Ts       = (float*)p; p += (size_t)CHN * 25 * sizeof(float);
    float* msum     = (float*)p; p += (size_t)NN * SN * sizeof(float);
    float* sbufA    = (float*)p; p += (size_t)NN * SN * sizeof(float);
    int*   char_idx = (int*)p;   p += (size_t)NN * sizeof(int);
    int*   sorted   = (int*)p;   p += (size_t)NN * sizeof(int);
    int*   bin_cnt  = (int*)p;   p += (size_t)CHN * sizeof(int);
    int*   bin_off  = (int*)p;   p += (size_t)CHN * sizeof(int);
    int*   bin_cur  = (int*)p;   p += (size_t)CHN * sizeof(int);

    softmax_k<<<(CHN * SN + 255) / 256, 256, 0, stream>>>(T, Ts);

    const float* cur = s0;
    for (int it = 0; it < ITERS; ++it) {
        float* nxt = (it & 1) ? sbufA : out;   // odd count => final lands in d_out
        hipMemsetAsync(msum, 0, (size_t)NN * SN * sizeof(float), stream);
        hipMemsetAsync(bin_cnt, 0, (size_t)CHN * sizeof(int), stream);
        scatter_k<<<(NE + 255) / 256, 256, 0, stream>>>(src, dst, cur, msum);
        char_k<<<(NN + 255) / 256, 256, 0, stream>>>(msum, char_idx, bin_cnt);
        scan_k<<<1, CHN, 0, stream>>>(bin_cnt, bin_off, bin_cur);
        compact_k<<<(NN + 255) / 256, 256, 0, stream>>>(char_idx, bin_off, bin_cur, sorted);
        update_wmma_k<<<CHN, 256, 0, stream>>>(cur, Ts, sorted, bin_off, bin_cnt, nxt);
        cur = nxt;
    }
}